// VolumeRenderer_88433376625124
// MI455X (gfx1250) — compile-verified
//
#include <hip/hip_runtime.h>

typedef __attribute__((ext_vector_type(2))) float v2f;
typedef __attribute__((ext_vector_type(8))) float v8f;

namespace {
constexpr int   kSamples      = 128;
constexpr int   kRaysPerWave  = 16;
constexpr int   kWavesPerBlock= 4;
constexpr int   kRaysPerBlock = kRaysPerWave * kWavesPerBlock;  // 64
constexpr int   kLdsStride    = 132;      // floats; rows 16B-aligned, conflict-free reads
constexpr float kFarDelta     = 1e10f;
}

__global__ __launch_bounds__(kWavesPerBlock * 32)
void volume_render_wmma(const float* __restrict__ rgb,
                        const float* __restrict__ density,
                        const float* __restrict__ dist,
                        float* __restrict__ out) {
  __shared__ float sig[kWavesPerBlock][kRaysPerWave][kLdsStride];

  const int lane = threadIdx.x & 31;
  const int wave = threadIdx.x >> 5;
  const int R    = (int)blockIdx.x * kRaysPerBlock + wave * kRaysPerWave;

  // ---- Phase A: coalesced b128 loads; sigma = density * delta -> LDS (one row per ray)
  for (int r = 0; r < kRaysPerWave; ++r) {
    const size_t base = (size_t)(R + r) * kSamples + lane * 4;
    const float4 d4 = *reinterpret_cast<const float4*>(dist    + base);
    const float4 q4 = *reinterpret_cast<const float4*>(density + base);
    const float dnext = __shfl_down(d4.x, 1, 32);          // first dist of next lane's chunk
    float4 sv;
    sv.x = q4.x * (d4.y - d4.x);
    sv.y = q4.y * (d4.z - d4.y);
    sv.z = q4.z * (d4.w - d4.z);
    sv.w = q4.w * ((lane == 31) ? kFarDelta : (dnext - d4.w));  // sample 127: delta = 1e10
    *reinterpret_cast<float4*>(&sig[wave][r][lane * 4]) = sv;
  }
  __syncthreads();

  // ---- Phase B: cumsum-as-matmul. D = L16(16x16 lower-tri ones) x SigmaSeg + carry.
  // Lane roles (wave32 WMMA layouts): lane%16 = ray (N); hi = lane>>4 selects the
  // K/M sub-rows each half-wave owns in the A/B/D register layouts.
  const int nray = lane & 15;
  const int hi   = lane >> 4;
  const float* __restrict__ srow = &sig[wave][nray][0];

  // A = lower-triangular ones, chunked K=4. A layout: VGPR0 holds K, K+2; VGPR1 K+1, K+3.
  v2f a[4];
#pragma unroll
  for (int ch = 0; ch < 4; ++ch) {
    const int kk = 4 * ch + 2 * hi;
    a[ch][0] = (kk     <= nray) ? 1.0f : 0.0f;
    a[ch][1] = (kk + 1 <= nray) ? 1.0f : 0.0f;
  }

  float off  = 0.0f;                       // inter-segment scan carry (per ray)
  float acc0 = 0.0f, acc1 = 0.0f, acc2 = 0.0f;

  for (int seg = 0; seg < 8; ++seg) {
    const int sbase = seg * 16;
    v8f c = {off, off, off, off, off, off, off, off};   // C[m][n] = carry[n] for all m
#pragma unroll
    for (int ch = 0; ch < 4; ++ch) {
      const v2f b = *reinterpret_cast<const v2f*>(srow + sbase + 4 * ch + 2 * hi);
      c = __builtin_amdgcn_wmma_f32_16x16x4_f32(false, a[ch], false, b,
                                                (short)0, c, false, false);
    }
    // c[v] = inclusive cumsum at sample sbase + v + 8*hi for ray R+nray.
    // Segment total = row m=15 -> D VGPR7 on hi lanes; broadcast to both halves.
    const float segtot = __shfl(c[7], 16 + nray, 32);

#pragma unroll
    for (int v = 0; v < 8; ++v) {
      const int   s    = sbase + v + 8 * hi;
      const float csum = c[v];
      const float sgm  = srow[s];
      // w = T*alpha = exp(-(csum - sigma)) - exp(-csum)
      const float w = __expf(sgm - csum) - __expf(-csum);
      const float* p = rgb + ((size_t)(R + nray) * kSamples + s) * 3;
      acc0 = fmaf(w, p[0], acc0);
      acc1 = fmaf(w, p[1], acc1);
      acc2 = fmaf(w, p[2], acc2);
    }
    off = segtot;
  }

  // Combine the two half-wave partials (samples m=0..7 vs m=8..15 of each segment).
  acc0 += __shfl_xor(acc0, 16, 32);
  acc1 += __shfl_xor(acc1, 16, 32);
  acc2 += __shfl_xor(acc2, 16, 32);
  if (hi == 0) {
    float* o = out + (size_t)(R + nray) * 3;   // coalesced 192B per wave
    o[0] = acc0; o[1] = acc1; o[2] = acc2;
  }
}

extern "C" void kernel_launch(void* const* d_in, const int* in_sizes, int n_in,
                              void* d_out, int out_size, void* d_ws, size_t ws_size,
                              hipStream_t stream) {
  const float* rgb     = (const float*)d_in[0];
  const float* density = (const float*)d_in[1];
  const float* dist    = (const float*)d_in[2];
  float* out = (float*)d_out;
  (void)d_ws; (void)ws_size; (void)n_in; (void)out_size;

  const int n_rays = in_sizes[2] / kSamples;        // 262144
  const int blocks = n_rays / kRaysPerBlock;        // 4096
  volume_render_wmma<<<blocks, kWavesPerBlock * 32, 0, stream>>>(rgb, density, dist, out);
}